// VCM_Decoder_23321672417650
// MI455X (gfx1250) — compile-verified
//
#include <hip/hip_runtime.h>

// ---------------------------------------------------------------------------
// WMMA fragment types (wave32):
//   A 16x4 f32  -> 64 vals / 32 lanes = 2 VGPRs -> v2f
//   B 4x16 f32  -> 2 VGPRs -> v2f
//   C/D 16x16   -> 8 VGPRs -> v8f
// ---------------------------------------------------------------------------
typedef __attribute__((ext_vector_type(2))) float v2f;
typedef __attribute__((ext_vector_type(8))) float v8f;

#define TM 64      // block tile rows
#define TN 64      // block tile cols
#define TKK 16     // K step per LDS stage
#define APAD 18    // A tile row stride (floats), even -> aligned b64 frag loads
#define TNP 80     // W tile col-pad: kp row stride = 160 floats -> bank-safe

// C[M,N] = A[M,K] * W[K,N] + bias[N]
// Requirements: M % 64 == 0, K % 16 == 0, N % 4 == 0 (edge tiles in N handled).
__global__ __launch_bounds__(128)
void gemm_wmma_f32_kernel(const float* __restrict__ A, const float* __restrict__ W,
                          const float* __restrict__ bias, float* __restrict__ C,
                          int M, int N, int K)
{
    // double-buffered tiles
    __shared__ float sA[2][TM][APAD];           // 2 * 64*18*4  = 9216 B
    __shared__ float sW[2][TKK / 2][TNP][2];    // 2 * 8*80*2*4 = 10240 B

    const int tid  = threadIdx.x;
    const int wave = tid >> 5;     // 0..3 -> 16-row strip within the 64-row tile
    const int lane = tid & 31;
    const int half = lane >> 4;    // 0/1: K split for A/B frags, M split for C/D
    const int l16  = lane & 15;
    const int m0 = blockIdx.y * TM;
    const int n0 = blockIdx.x * TN;

    v8f acc[4];
#pragma unroll
    for (int j = 0; j < 4; ++j)
        acc[j] = (v8f){0.f,0.f,0.f,0.f,0.f,0.f,0.f,0.f};

    // stage one 64x16 A tile and 16x64 W tile into LDS buffer `buf`
    auto stage = [&](int buf, int kb) {
#pragma unroll
        for (int i = tid; i < (TM * TKK) / 4; i += 128) {   // 256 float4
            const int r  = i >> 2;                          // 4 float4 per row
            const int c4 = (i & 3) * 4;
            const float4 v = *(const float4*)(A + (size_t)(m0 + r) * K + kb + c4);
            sA[buf][r][c4 + 0] = v.x; sA[buf][r][c4 + 1] = v.y;
            sA[buf][r][c4 + 2] = v.z; sA[buf][r][c4 + 3] = v.w;
        }
#pragma unroll
        for (int i = tid; i < (TKK * TN) / 4; i += 128) {   // 256 float4
            const int r  = i >> 4;                          // 16 float4 per row
            const int c4 = (i & 15) * 4;
            const int gn = n0 + c4;
            float4 v = make_float4(0.f, 0.f, 0.f, 0.f);
            if (gn < N)     // N % 4 == 0 -> float4 fully in or out of range
                v = *(const float4*)(W + (size_t)(kb + r) * N + gn);
            const int kp = r >> 1, ko = r & 1;              // pair-interleaved
            sW[buf][kp][c4 + 0][ko] = v.x; sW[buf][kp][c4 + 1][ko] = v.y;
            sW[buf][kp][c4 + 2][ko] = v.z; sW[buf][kp][c4 + 3][ko] = v.w;
        }
    };

    stage(0, 0);
    int buf = 0;
    const int arow = wave * 16 + l16;

    for (int kb = 0; kb < K; kb += TKK) {
        __syncthreads();   // current buf staged; prior compute on other buf done
        if (kb + TKK < K) {
            stage(buf ^ 1, kb + TKK);
            if (kb + 2 * TKK < K) {
                // L2 prefetch two K-tiles ahead (global_prefetch_b8)
                __builtin_prefetch(A + (size_t)(m0 + (tid & 63)) * K + kb + 2 * TKK, 0, 0);
                const int pn = n0 + (tid >> 4) * 8;
                if (pn < N)
                    __builtin_prefetch(W + (size_t)(kb + 2 * TKK + (tid & 15)) * N + pn, 0, 0);
            }
        }

        // ---- compute: 4 K-steps of 4, 4 N sub-tiles; 16 WMMAs per wave ----
#pragma unroll
        for (int kk = 0; kk < TKK; kk += 4) {
            // A frag: lanes 0-15 hold K=kk+0,kk+1 ; lanes 16-31 hold K=kk+2,kk+3
            const v2f a = *(const v2f*)&sA[buf][arow][kk + half * 2];
#pragma unroll
            for (int j = 0; j < 4; ++j) {
                // B frag: one aligned 8B load (pair-interleaved layout)
                const v2f bf = *(const v2f*)&sW[buf][(kk >> 1) + half][j * 16 + l16][0];
                acc[j] = __builtin_amdgcn_wmma_f32_16x16x4_f32(
                    /*neg_a=*/false, a, /*neg_b=*/false, bf,
                    /*c_mod=*/(short)0, acc[j],
                    /*reuse_a=*/false, /*reuse_b=*/false);
            }
        }
        buf ^= 1;
    }

    // ---- epilogue: D layout -> VGPR e: lanes0-15 row e, lanes16-31 row e+8 ----
    const int rbase = m0 + wave * 16 + half * 8;
#pragma unroll
    for (int j = 0; j < 4; ++j) {
        const int col = n0 + j * 16 + l16;
        if (col < N) {
            const float bv = bias[col];
#pragma unroll
            for (int e = 0; e < 8; ++e)
                C[(size_t)(rbase + e) * N + col] = acc[j][e] + bv;
        }
    }
}

// ---------------------------------------------------------------------------
// coverage: unc[b][n] = (sum_t mask[b][t][n]) == 0
// ---------------------------------------------------------------------------
__global__ void coverage_kernel(const unsigned char* __restrict__ mask,
                                unsigned char* __restrict__ unc, int T, int N)
{
    const int n = blockIdx.x * blockDim.x + threadIdx.x;
    const int b = blockIdx.y;
    if (n >= N) return;
    const unsigned char* p = mask + ((size_t)b * T) * N + n;
    int s = 0;
    for (int t = 0; t < T; ++t) s |= p[(size_t)t * N];
    unc[(size_t)b * N + n] = (unsigned char)(s == 0);
}

// ---------------------------------------------------------------------------
// build_map: per sample b, map[n] = k         if n == index[b][k]
//                         = K + rank(n)       if uncovered rest col, rank < R
//                         = -1                otherwise (output 0)
// One block per b; scatter then chunked block scan (in-place in map).
// ---------------------------------------------------------------------------
__global__ __launch_bounds__(256)
void build_map_kernel(const int* __restrict__ index, const unsigned char* __restrict__ unc,
                      int* __restrict__ map, int N, int K, int R)
{
    const int b = blockIdx.x;
    const int t = threadIdx.x;
    __shared__ int s_scan[256];
    __shared__ int s_carry;

    int* pm = map + (size_t)b * N;
    for (int n = t; n < N; n += 256) pm[n] = -1;
    __syncthreads();
    const int* ib = index + (size_t)b * K;
    for (int k = t; k < K; k += 256) pm[ib[k]] = k;     // indices are distinct
    if (t == 0) s_carry = 0;
    __syncthreads();

    for (int base = 0; base < N; base += 256) {
        const int n   = base + t;
        const int pos = (n < N) ? pm[n] : -1;
        const int u   = (n < N) ? (int)unc[(size_t)b * N + n] : 0;
        const int flag = (u && pos < 0) ? 1 : 0;
        s_scan[t] = flag;
        __syncthreads();
#pragma unroll
        for (int off = 1; off < 256; off <<= 1) {
            const int add = (t >= off) ? s_scan[t - off] : 0;
            __syncthreads();
            s_scan[t] += add;
            __syncthreads();
        }
        const int rank = s_carry + s_scan[t] - flag;    // exclusive prefix among rest cols
        if (n < N) {
            int v;
            if (pos >= 0)               v = pos;
            else if (flag && rank < R)  v = K + rank;
            else                        v = -1;
            pm[n] = v;
        }
        __syncthreads();
        if (t == 0) s_carry += s_scan[255];
        __syncthreads();
    }
}

// ---------------------------------------------------------------------------
// assemble: out[b][t][n] from h2 / x_rest / zero per map
// ---------------------------------------------------------------------------
__global__ __launch_bounds__(256)
void assemble_kernel(const int* __restrict__ map, const float* __restrict__ h2,
                     const float* __restrict__ xr, float* __restrict__ out,
                     int T, int N, int K, int R, int total)
{
    const int idx = blockIdx.x * 256 + threadIdx.x;
    if (idx >= total) return;
    const int n  = idx % N;
    const int bt = idx / N;
    const int b  = bt / T;
    const int m  = map[(size_t)b * N + n];
    float v = 0.f;
    if (m >= 0)
        v = (m < K) ? h2[(size_t)bt * K + m] : xr[(size_t)bt * R + (m - K)];
    out[idx] = v;
}

// ---------------------------------------------------------------------------
extern "C" void kernel_launch(void* const* d_in, const int* in_sizes, int n_in,
                              void* d_out, int out_size, void* d_ws, size_t ws_size,
                              hipStream_t stream)
{
    (void)in_sizes; (void)n_in; (void)out_size; (void)ws_size;
    const float*         x     = (const float*)d_in[0];
    const unsigned char* mask  = (const unsigned char*)d_in[1];
    const int*           index = (const int*)d_in[2];
    const float*         W1    = (const float*)d_in[3];
    const float*         b1    = (const float*)d_in[4];
    const float*         W2    = (const float*)d_in[5];
    const float*         b2    = (const float*)d_in[6];
    const float*         W3    = (const float*)d_in[7];
    const float*         b3    = (const float*)d_in[8];
    float* out = (float*)d_out;

    const int Bb = 16, T = 256, IN = 800, K = 4000, R = 1000, N = 20000;
    const int M  = Bb * T;                     // 4096

    char* ws = (char*)d_ws;
    float* h1    = (float*)(ws);                                   // M*K f32
    float* h2    = (float*)(ws + (size_t)M * K * 4);               // M*K f32
    float* xrest = (float*)(ws);                                   // alias h1 (M*R f32)
    int*   map   = (int*)  (ws + (size_t)2 * M * K * 4);           // Bb*N int
    unsigned char* unc = (unsigned char*)(ws + (size_t)2 * M * K * 4 + (size_t)Bb * N * 4);

    const dim3 blk(128);
    // h1 = x @ W1 + b1           [4096 x 4000], K=800
    gemm_wmma_f32_kernel<<<dim3((K + TN - 1) / TN, M / TM), blk, 0, stream>>>(
        x, W1, b1, h1, M, K, IN);
    // h2 = h1 @ W2 + b2          [4096 x 4000], K=4000
    gemm_wmma_f32_kernel<<<dim3((K + TN - 1) / TN, M / TM), blk, 0, stream>>>(
        h1, W2, b2, h2, M, K, K);
    // x_rest = h2 @ W3 + b3      [4096 x 1000], K=4000
    gemm_wmma_f32_kernel<<<dim3((R + TN - 1) / TN, M / TM), blk, 0, stream>>>(
        h2, W3, b3, xrest, M, R, K);

    coverage_kernel<<<dim3((N + 255) / 256, Bb), 256, 0, stream>>>(mask, unc, T, N);
    build_map_kernel<<<Bb, 256, 0, stream>>>(index, unc, map, N, K, R);

    const int total = M * N;   // 81,920,000
    assemble_kernel<<<(total + 255) / 256, 256, 0, stream>>>(
        map, h2, xrest, out, T, N, K, R, total);
}